// Attention_48911087567198
// MI455X (gfx1250) — compile-verified
//
#include <hip/hip_runtime.h>
#include <math.h>

typedef float v2f __attribute__((ext_vector_type(2)));
typedef float v8f __attribute__((ext_vector_type(8)));

#define Bb 64
#define Ss 2048
#define Hh 1024
#define Kk 2048  // = 2*H

// ---------------------------------------------------------------------------
// Kernel 1: energy[64,1024] = cs[64,2048] @ W[1024,2048]^T + bias[1024]
// WMMA f32 16x16x4. Block = 128 threads (4 waves); wave w -> M-tile w (0..3);
// blockIdx.x -> N-tile (0..63). K loop: 2048 in steps of 8 (2 wmma / iter).
// ---------------------------------------------------------------------------
__global__ void energy_gemm_wmma(const float* __restrict__ cs,
                                 const float* __restrict__ W,
                                 const float* __restrict__ bias,
                                 float* __restrict__ energy)
{
    const int lane   = threadIdx.x & 31;
    const int m_tile = threadIdx.x >> 5;   // 0..3
    const int n_tile = blockIdx.x;         // 0..63
    const int nlo    = lane & 15;          // 0..15
    const int khalf  = (lane >> 4) << 1;   // 0 or 2 (ISA A/B f32 layout)

    // A fragment source: row M = m_tile*16 + nlo of cs
    const float* a_row = cs + (size_t)(m_tile * 16 + nlo) * Kk + khalf;
    // B = W^T: lane feeds column n = n_tile*16 + nlo, i.e. row n of W
    const float* b_row = W  + (size_t)(n_tile * 16 + nlo) * Kk + khalf;

    const float bv = bias[n_tile * 16 + nlo];   // bias depends only on N
    v8f c0, c1;
#pragma unroll
    for (int i = 0; i < 8; ++i) { c0[i] = bv; c1[i] = 0.0f; }

    for (int k = 0; k < Kk; k += 8) {
        v2f a0 = *(const v2f*)(a_row + k);
        v2f b0 = *(const v2f*)(b_row + k);
        v2f a1 = *(const v2f*)(a_row + k + 4);
        v2f b1 = *(const v2f*)(b_row + k + 4);
        c0 = __builtin_amdgcn_wmma_f32_16x16x4_f32(false, a0, false, b0,
                                                   (short)0, c0, false, false);
        c1 = __builtin_amdgcn_wmma_f32_16x16x4_f32(false, a1, false, b1,
                                                   (short)0, c1, false, false);
    }
    v8f c = c0 + c1;

    // D layout: row M = i + (lane>>4)*8, col N = lane&15
    const int mbase = m_tile * 16 + ((lane >> 4) << 3);
    const int ncol  = n_tile * 16 + nlo;
#pragma unroll
    for (int i = 0; i < 8; ++i)
        energy[(size_t)(mbase + i) * Hh + ncol] = c[i];
}

// ---------------------------------------------------------------------------
// Kernel 2 (bandwidth-bound phase): attn[b,s] = dot(enc[b,s,:], energy[b,:])
// One wave per (b,s) row; 8 float4 loads per lane, fully coalesced b128.
// ---------------------------------------------------------------------------
__global__ void attn_dot(const float* __restrict__ enc,
                         const float* __restrict__ energy,
                         float* __restrict__ attn)
{
    const int wave = threadIdx.x >> 5;
    const int lane = threadIdx.x & 31;
    const long long idx = (long long)blockIdx.x * 8 + wave;   // 0 .. B*S-1
    const int b = (int)(idx >> 11);          // / 2048
    const float* row = enc    + (size_t)idx * Hh;
    const float* e   = energy + (size_t)b   * Hh;

    float acc = 0.0f;
#pragma unroll
    for (int i = 0; i < 8; ++i) {
        const float4 r  = *(const float4*)(row + i * 128 + lane * 4);
        const float4 ev = *(const float4*)(e   + i * 128 + lane * 4);
        acc = fmaf(r.x, ev.x, acc);
        acc = fmaf(r.y, ev.y, acc);
        acc = fmaf(r.z, ev.z, acc);
        acc = fmaf(r.w, ev.w, acc);
    }
#pragma unroll
    for (int off = 16; off > 0; off >>= 1)
        acc += __shfl_xor(acc, off, 32);
    if (lane == 0) attn[idx] = acc;
}

// ---------------------------------------------------------------------------
// Kernel 3: in-place softmax over each row of 2048. One 256-thread block/row.
// ---------------------------------------------------------------------------
__global__ void softmax2048(float* __restrict__ x)
{
    __shared__ float red[8];
    float* row = x + (size_t)blockIdx.x * Ss;
    const int t = threadIdx.x;            // 0..255

    float v[8];
    float m = -INFINITY;
#pragma unroll
    for (int i = 0; i < 8; ++i) {
        v[i] = row[t + i * 256];
        m = fmaxf(m, v[i]);
    }
#pragma unroll
    for (int off = 16; off > 0; off >>= 1)
        m = fmaxf(m, __shfl_xor(m, off, 32));
    if ((t & 31) == 0) red[t >> 5] = m;
    __syncthreads();
    m = red[0];
#pragma unroll
    for (int i = 1; i < 8; ++i) m = fmaxf(m, red[i]);

    float s = 0.0f;
#pragma unroll
    for (int i = 0; i < 8; ++i) {
        v[i] = __expf(v[i] - m);
        s += v[i];
    }
#pragma unroll
    for (int off = 16; off > 0; off >>= 1)
        s += __shfl_xor(s, off, 32);
    __syncthreads();                       // red[] reuse
    if ((t & 31) == 0) red[t >> 5] = s;
    __syncthreads();
    s = 0.0f;
#pragma unroll
    for (int i = 0; i < 8; ++i) s += red[i];
    const float inv = 1.0f / s;
#pragma unroll
    for (int i = 0; i < 8; ++i)
        row[t + i * 256] = v[i] * inv;
}

// ---------------------------------------------------------------------------
extern "C" void kernel_launch(void* const* d_in, const int* in_sizes, int n_in,
                              void* d_out, int out_size, void* d_ws, size_t ws_size,
                              hipStream_t stream)
{
    const float* cs   = (const float*)d_in[0];  // [64, 2048]
    const float* enc  = (const float*)d_in[1];  // [64, 2048, 1024]
    const float* W    = (const float*)d_in[2];  // [1024, 2048]
    const float* bias = (const float*)d_in[3];  // [1024]
    float*       out  = (float*)d_out;          // [64, 1, 2048]
    float*       energy = (float*)d_ws;         // 64*1024 floats (256 KB)

    // Phase 1: WMMA GEMM -> energy
    energy_gemm_wmma<<<64, 128, 0, stream>>>(cs, W, bias, energy);
    // Phase 2: bandwidth-bound batched GEMV -> out (raw logits)
    attn_dot<<<(Bb * Ss) / 8, 256, 0, stream>>>(enc, energy, out);
    // Phase 3: in-place softmax over S
    softmax2048<<<Bb, 256, 0, stream>>>(out);
}